// NeighborRadiusSearchLayer_55490977465087
// MI455X (gfx1250) — compile-verified
//
#include <hip/hip_runtime.h>
#include <hip/hip_bf16.h>

// ---------------------------------------------------------------------------
// Fixed-radius neighbor search on gfx1250 (MI455X).
//
//   d2(q,i) = |q|^2 - 2 q.i + |i|^2
// One V_WMMA_F32_16X16X4_F32 computes a 16(input) x 16(query) tile of
// (d2 - |q|^2):   A row m = [ix, iy, iz, |i|^2],  B col q = [-2qx,-2qy,-2qz, 1]
//
// C/D layout (wave32): lane owns query column q = lane%16; VGPR j holds input
// row j (lanes 0-15) or j+8 (lanes 16-31).  So the two lane-halves of a query
// each see 8 candidates per WMMA and keep private LDS hit-lists (no atomics).
//
// Inner-loop structure: software-pipelined ds_load of the next A-fragment,
// and the rare candidate-processing block is guarded by a v_min_num tree +
// single v_cmpx (P(tile has any hit) ~ 42%, so most iterations skip it).
// ---------------------------------------------------------------------------

typedef __attribute__((ext_vector_type(2))) float v2f;
typedef __attribute__((ext_vector_type(8))) float v8f;

#define NPTS   16384
#define R2     0.0064f          /* 0.08^2 */
#define KMAX   64               /* MAX_NEIGHBORS */
#define CAP    64               /* per (query,half) candidate capacity; E[hits]=~17.5 */
#define CHUNK  4096             /* inputs staged in LDS per round (64 KB) */
#define TILES  (CHUNK / 16)
#define WAVES  4
#define BLOCK  (WAVES * 32)     /* 128 threads */
#define QPB    (WAVES * 16)     /* 64 queries per block */

__global__ void nbr_radius_wmma_kernel(const float* __restrict__ inp,
                                       const float* __restrict__ qry,
                                       int* __restrict__ nbr,
                                       int* __restrict__ rs)
{
    extern __shared__ char smem[];
    float4*             stage  = reinterpret_cast<float4*>(smem);              // [CHUNK] x,y,z,|p|^2
    const float2*       stage2 = reinterpret_cast<const float2*>(smem);
    unsigned long long* cand   = reinterpret_cast<unsigned long long*>(smem + CHUNK * 16);

    const int tid  = threadIdx.x;
    const int wave = tid >> 5;
    const int lane = tid & 31;
    const int q    = lane & 15;   // query column owned by this lane
    const int half = lane >> 4;   // which 8 candidate rows this lane sees

    // ---- per-lane query setup (B operand) ---------------------------------
    const int qglob = blockIdx.x * QPB + wave * 16 + q;
    const float qx = qry[qglob * 3 + 0];
    const float qy = qry[qglob * 3 + 1];
    const float qz = qry[qglob * 3 + 2];
    const float qn2 = qx * qx + qy * qy + qz * qz;
    const float thr = R2 - qn2;            // D <= thr  <=>  d2 <= r^2

    v2f b;                                  // B 4x16: v0={K0|K2}, v1={K1|K3}
    b.x = half ? (-2.0f * qz) : (-2.0f * qx);
    b.y = half ? 1.0f         : (-2.0f * qy);

    unsigned long long* mylist = cand + (((wave * 16 + q) * 2 + half) * CAP);
    int cntW = 0;                           // within-radius count (uncapped)
    int cntS = 0;                           // stored count (<= CAP)

    // ---- brute-force scan: LDS-staged input chunks ------------------------
    for (int cbase = 0; cbase < NPTS; cbase += CHUNK) {
        for (int i = tid; i < CHUNK; i += BLOCK) {
            const int g = cbase + i;
            const float x = inp[g * 3 + 0];
            const float y = inp[g * 3 + 1];
            const float z = inp[g * 3 + 2];
            float4 v; v.x = x; v.y = y; v.z = z; v.w = x * x + y * y + z * z;
            stage[i] = v;
        }
        __syncthreads();

        // software-pipelined A-fragment loads (LDS latency hides under WMMA)
        float2 av = stage2[q * 2 + half];   // tile 0
#pragma unroll 2
        for (int t = 0; t < TILES; ++t) {
            v2f a; a.x = av.x; a.y = av.y;
            const int tn = (t + 1 < TILES) ? t + 1 : t;
            const float2 avn = stage2[(tn * 16 + q) * 2 + half];

            v8f c = {};
            c = __builtin_amdgcn_wmma_f32_16x16x4_f32(
                    false, a, false, b, (short)0, c, false, false);

            // cheap wave-level guard: min of the 8 distances vs threshold
            const float m01 = fminf(c[0], c[1]);
            const float m23 = fminf(c[2], c[3]);
            const float m45 = fminf(c[4], c[5]);
            const float m67 = fminf(c[6], c[7]);
            const float mall = fminf(fminf(m01, m23), fminf(m45, m67));

            if (mall <= thr) {              // rare: ~42% of tiles wave-wide
                const int gtile = cbase + t * 16 + half * 8;
#pragma unroll
                for (int j = 0; j < 8; ++j) {
                    const float D = c[j];   // d2 - |q|^2 for input row gtile+j
                    if (D <= thr) {
                        const float d2 = fmaxf(D + qn2, 0.0f);
                        const unsigned long long key =
                            ((unsigned long long)__float_as_uint(d2) << 32) |
                            (unsigned)(gtile + j);   // stable tie-break: lower index first
                        if (cntS < CAP) { mylist[cntS] = key; cntS++; }
                        cntW++;
                    }
                }
            }
            av = avn;
        }
        __syncthreads();
    }

    // ---- merge the two half-lists per query, nearest first ----------------
    const int cntW_hi = __shfl(cntW, lane + 16, 32);
    const int cntS_hi = __shfl(cntS, lane + 16, 32);
    __syncthreads();

    if (lane < 16) {
        unsigned long long* l0 = cand + (((wave * 16 + q) * 2 + 0) * CAP);
        unsigned long long* l1 = cand + (((wave * 16 + q) * 2 + 1) * CAP);
        const int n0 = cntS, n1 = cntS_hi;
        const int cnt_out = min(cntW + cntW_hi, KMAX);
        const int kout    = min(cnt_out, n0 + n1);

        int* row = nbr + (size_t)qglob * KMAX;
        for (int k = 0; k < kout; ++k) {
            unsigned long long best = ~0ull; int bi = 0, bl = 0;
            for (int i = 0; i < n0; ++i) if (l0[i] < best) { best = l0[i]; bi = i; bl = 0; }
            for (int i = 0; i < n1; ++i) if (l1[i] < best) { best = l1[i]; bi = i; bl = 1; }
            row[k] = (int)(unsigned)(best & 0xffffffffu);
            if (bl) l1[bi] = ~0ull; else l0[bi] = ~0ull;
        }
        for (int k = kout; k < KMAX; ++k) row[k] = -1;
        rs[qglob + 1] = cnt_out;            // raw count; kernel 2 turns into cumsum
    }
}

// ---- row_splits prefix sum: 256 threads x 64 counts each -------------------
__global__ void row_splits_scan_kernel(int* __restrict__ rs)
{
    __shared__ int sums[256];
    const int t    = threadIdx.x;
    const int base = 1 + t * 64;

    int v[64];
    int acc = 0;
#pragma unroll
    for (int i = 0; i < 64; ++i) { acc += rs[base + i]; v[i] = acc; }
    sums[t] = acc;
    __syncthreads();

    if (t == 0) {
        int run = 0;
        for (int i = 0; i < 256; ++i) { const int s = sums[i]; sums[i] = run; run += s; }
    }
    __syncthreads();

    const int off = sums[t];
#pragma unroll
    for (int i = 0; i < 64; ++i) rs[base + i] = v[i] + off;
    if (t == 0) rs[0] = 0;
}

extern "C" void kernel_launch(void* const* d_in, const int* in_sizes, int n_in,
                              void* d_out, int out_size, void* d_ws, size_t ws_size,
                              hipStream_t stream)
{
    const float* inp = (const float*)d_in[0];   // inp_positions [16384,3]
    const float* qry = (const float*)d_in[1];   // out_positions [16384,3]
    int* out = (int*)d_out;                     // int32 outputs (bit-compatible view)
    int* nbr = out;                             // [16384, 64]
    int* rs  = out + NPTS * KMAX;               // [16385]

    const size_t shmem = (size_t)CHUNK * 16 + (size_t)QPB * 2 * CAP * 8;  // 128 KB LDS

    nbr_radius_wmma_kernel<<<NPTS / QPB, BLOCK, shmem, stream>>>(inp, qry, nbr, rs);
    row_splits_scan_kernel<<<1, 256, 0, stream>>>(rs);
}